// TtMistralAttention_12146167513754
// MI455X (gfx1250) — compile-verified
//
#include <hip/hip_runtime.h>
#include <hip/hip_bf16.h>
#include <math.h>

// Problem constants (match reference)
#define BATCH  32
#define NHEAD  32
#define NKVH   8
#define HDIM   128
#define SWIN   4096
#define DMODEL 4096
#define GQ     4                      // NH / NKV
#define QKVN   (NHEAD*HDIM + 2*NKVH*HDIM)   // 6144

typedef float v2f __attribute__((ext_vector_type(2)));
typedef float v8f __attribute__((ext_vector_type(8)));
typedef int   v4i __attribute__((ext_vector_type(4)));

// CDNA5 async global->LDS path (ASYNCcnt-tracked), guarded so both toolchains compile
#if defined(__has_builtin)
#  if __has_builtin(__builtin_amdgcn_global_load_async_to_lds_b128) && \
      __has_builtin(__builtin_amdgcn_s_wait_asynccnt)
#    define USE_ASYNC_LDS 1
#  endif
#endif
#ifndef USE_ASYNC_LDS
#  define USE_ASYNC_LDS 0
#endif

#if USE_ASYNC_LDS
typedef __attribute__((address_space(1))) v4i gbl_v4i;   // global int4
typedef __attribute__((address_space(3))) v4i lds_v4i;   // LDS int4
__device__ __forceinline__ void async_row_b128(const float* gsrc, float* ldst) {
  // one wave-wide instruction: 32 lanes x 16B = 512B (one K/V row)
  __builtin_amdgcn_global_load_async_to_lds_b128((gbl_v4i*)gsrc, (lds_v4i*)ldst, 0, 0);
}
#endif

// D(16x16,f32) = A(16x4,f32) * B(4x16,f32) + C   (layouts per ISA 7.12.2)
__device__ __forceinline__ v8f wmma4(v2f a, v2f b, v8f c) {
  return __builtin_amdgcn_wmma_f32_16x16x4_f32(
      /*neg_a=*/false, a, /*neg_b=*/false, b,
      /*c_mod=*/(short)0, c, /*reuse_a=*/false, /*reuse_b=*/false);
}

// ---------------------------------------------------------------------------
// Batched wave-tile fp32 WMMA GEMM:  C[y] = A[y] * B.
// Each wave owns one 16-wide N column strip and MT row tiles (B streamed once).
// Pointer-stepped addressing: no per-iteration 64-bit multiplies.
// ---------------------------------------------------------------------------
template<int MT>
__global__ __launch_bounds__(128)
void gemm_f32_wmma(const float* __restrict__ Abase, long strideA,
                   const float* __restrict__ B,
                   float* __restrict__ Cbase, long strideC,
                   int M, int N, int K, int lda, int ldb, int ldc)
{
  const float* A = Abase + (long)blockIdx.y * strideA;
  float*       C = Cbase + (long)blockIdx.y * strideC;
  const int wave = threadIdx.x >> 5;
  const int lane = threadIdx.x & 31;
  const int m    = lane & 15;
  const int half = lane >> 4;
  const int nt   = blockIdx.x * (blockDim.x >> 5) + wave;
  const int ncol = nt * 16 + m;

  const float* ap[MT];
#pragma unroll
  for (int i = 0; i < MT; ++i) {
    const int rowA = i * 16 + m;
    ap[i] = A + (long)(rowA < M ? rowA : 0) * lda + 2 * half;
  }
  const float* bp0 = B + (long)(2 * half) * ldb + ncol;   // row kr
  const float* bp1 = bp0 + ldb;                           // row kr+1
  const long   step   = 4L * (long)ldb;
  const long   pfdist = 256L * (long)ldb;                 // 256 K-rows ahead

  v8f acc[MT];
#pragma unroll
  for (int i = 0; i < MT; ++i) acc[i] = (v8f){0.f,0.f,0.f,0.f,0.f,0.f,0.f,0.f};

#pragma unroll 4
  for (int k = 0; k < K; k += 4) {
    if (((k >> 2) & 15) == 0 && (k + 260) < K)
      __builtin_prefetch(bp0 + pfdist, 0, 0);             // global_prefetch_b8
    v2f b;
    b.x = *bp0;
    b.y = *bp1;
#pragma unroll
    for (int i = 0; i < MT; ++i) {
      v2f a = *(const v2f*)(ap[i]);
      acc[i] = wmma4(a, b, acc[i]);
      ap[i] += 4;
    }
    bp0 += step;
    bp1 += step;
  }

#pragma unroll
  for (int i = 0; i < MT; ++i)
#pragma unroll
    for (int j = 0; j < 8; ++j) {
      const int row = i * 16 + j + 8 * half;
      if (row < M) C[(long)row * ldc + ncol] = acc[i][j];
    }
}

// ---------------------------------------------------------------------------
// Flash-style GQA attention over the full 4096 window.
// One workgroup (8 waves) per (b, kv-head); 512-key stripe per wave; online
// softmax; WMMA for Q*K^T and P*V. Both K and V chunks are DMA'd into LDS via
// global_load_async_to_lds_b128 (one instruction per 512B row, ASYNCcnt) and
// read back through bank-conflict-free padded DS loads. s == current_pos
// reads k/v from the freshly projected vectors (inputs untouched).
// ---------------------------------------------------------------------------
#define KPAD (HDIM + 4)   // transposed K reads: stride 132 -> distinct banks
#define VPAD (HDIM + 8)   // paired V row reads: stride 136 -> halves on disjoint banks

__global__ __launch_bounds__(256)
void attn_flash_wmma(const float* __restrict__ qrot,
                     const float* __restrict__ krot,
                     const float* __restrict__ xqkv,   // for current v
                     const float* __restrict__ cache_k,
                     const float* __restrict__ cache_v,
                     const float* __restrict__ mask,
                     const int*   __restrict__ curpos_p,
                     float*       __restrict__ ctxout)
{
  const int bk   = blockIdx.x;
  const int b    = bk / NKVH;
  const int kv   = bk % NKVH;
  const int tid  = threadIdx.x;
  const int wave = tid >> 5;
  const int lane = tid & 31;
  const int m    = lane & 15;
  const int half = lane >> 4;
  const int cur  = *curpos_p;

#if USE_ASYNC_LDS
  __shared__ float lds_k[8][16 * KPAD];  // staged K chunks (~66KB)
  __shared__ float lds_v[8][16 * VPAD];  // staged V chunks (~68KB)
#endif
  __shared__ float lds_p[8][16][16];     // per-wave P tile relayout
  __shared__ float lds_ctx[8][GQ][HDIM]; // per-wave partial context
  __shared__ float lds_m[8][GQ];
  __shared__ float lds_l[8][GQ];

  const float scale = 0.08838834764831845f;  // 1/sqrt(128)

  // ---- load Q fragments (A-matrix layout), pre-scaled; rows >= GQ are zero
  v2f qa[32];
  {
    const int qrow = b * NHEAD + kv * GQ + (m < GQ ? m : 0);
    const float* qbase = qrot + (long)qrow * HDIM;
    const bool ok = (m < GQ);
#pragma unroll
    for (int t = 0; t < 32; ++t) {
      const int d = t * 4 + 2 * half;
      if (ok) { v2f q = *(const v2f*)(qbase + d); qa[t].x = q.x * scale; qa[t].y = q.y * scale; }
      else    { qa[t].x = 0.f; qa[t].y = 0.f; }
    }
  }

  const float* kbase = cache_k + (long)(b * NKVH + kv) * SWIN * HDIM;
  const float* vbase = cache_v + (long)(b * NKVH + kv) * SWIN * HDIM;
  const float* kcur  = krot + (long)(b * NKVH + kv) * HDIM;
  const float* vcur  = xqkv + (long)b * QKVN + (NHEAD * HDIM + NKVH * HDIM) + kv * HDIM;

  float mrow[8], lrow[8];
  v8f ctx[8];
#pragma unroll
  for (int i = 0; i < 8; ++i) { mrow[i] = -1e30f; lrow[i] = 0.f; }
#pragma unroll
  for (int i = 0; i < 8; ++i) ctx[i] = (v8f){0.f,0.f,0.f,0.f,0.f,0.f,0.f,0.f};

  const int s_base = wave * (SWIN / 8);          // 512-key stripe per wave
  for (int c = 0; c < (SWIN / 8) / 16; ++c) {    // 32 chunks of 16 keys
    const int s0 = s_base + c * 16;

#if USE_ASYNC_LDS
    // ---- DMA K and V chunks (16 rows x 512B each) into LDS
#pragma unroll
    for (int t = 0; t < 16; ++t) {
      const int s = s0 + t;
      const bool is_cur = (s == cur);
      const float* ksrc = (is_cur ? kcur : (kbase + (long)s * HDIM)) + lane * 4;
      const float* vsrc = (is_cur ? vcur : (vbase + (long)s * HDIM)) + lane * 4;
      async_row_b128(ksrc, &lds_k[wave][t * KPAD + lane * 4]);
      async_row_b128(vsrc, &lds_v[wave][t * VPAD + lane * 4]);
    }
    __builtin_amdgcn_s_wait_asynccnt(0);         // ASYNCcnt -> 0: chunk resident
    const float* krow_l = &lds_k[wave][m * KPAD];
#else
    const int keyn = s0 + m;
    const float* krow_l = (keyn == cur) ? kcur : (kbase + (long)keyn * HDIM);
#endif

    // ---- scores 16x16 = Q(16x128) * Kchunk^T   (32 WMMAs over HD)
    v8f sc = (v8f){0.f,0.f,0.f,0.f,0.f,0.f,0.f,0.f};
#pragma unroll
    for (int t = 0; t < 32; ++t) {
      const int d = t * 4 + 2 * half;
      v2f kb = *(const v2f*)(krow_l + d);
      sc = wmma4(qa[t], kb, sc);
    }

    // ---- online softmax per row (rows live across the 16 lanes of a half)
    float pval[8], alpha[8];
#pragma unroll
    for (int i = 0; i < 8; ++i) {
      const int g = i + 8 * half;
      float v = sc[i];
      if (g < GQ) v += mask[(long)((kv * GQ + g) * BATCH + b) * SWIN + s0 + m];
      float rmax = v;
      rmax = fmaxf(rmax, __shfl_xor(rmax, 1, 32));
      rmax = fmaxf(rmax, __shfl_xor(rmax, 2, 32));
      rmax = fmaxf(rmax, __shfl_xor(rmax, 4, 32));
      rmax = fmaxf(rmax, __shfl_xor(rmax, 8, 32));
      const float mnew = fmaxf(mrow[i], rmax);
      const float a = __expf(mrow[i] - mnew);
      const float p = __expf(v - mnew);
      float rsum = p;
      rsum += __shfl_xor(rsum, 1, 32);
      rsum += __shfl_xor(rsum, 2, 32);
      rsum += __shfl_xor(rsum, 4, 32);
      rsum += __shfl_xor(rsum, 8, 32);
      lrow[i] = lrow[i] * a + rsum;
      mrow[i] = mnew;
      alpha[i] = a;
      pval[i] = p;
    }

    // rescale running context by alpha (per output row)
#pragma unroll
    for (int n = 0; n < 8; ++n)
#pragma unroll
      for (int i = 0; i < 8; ++i) ctx[n][i] *= alpha[i];

    // ---- P (C-layout) -> LDS -> A-layout fragments (wave-private, DS in-order)
#pragma unroll
    for (int i = 0; i < 8; ++i) lds_p[wave][i + 8 * half][m] = pval[i];
    v2f pa[4];
#pragma unroll
    for (int t = 0; t < 4; ++t)
      pa[t] = *(const v2f*)&lds_p[wave][m][t * 4 + 2 * half];

    // ---- ctx(16x128) += P(16x16) * Vchunk(16x128)   (32 WMMAs)
    //      fragments hoisted so loads clause together before the WMMA burst
#pragma unroll
    for (int t = 0; t < 4; ++t) {
      v2f vb[8];
#if USE_ASYNC_LDS
      const float* vrow = &lds_v[wave][(t * 4 + 2 * half) * VPAD];  // rows k0, k0+1
#pragma unroll
      for (int n = 0; n < 8; ++n) {
        vb[n].x = vrow[n * 16 + m];
        vb[n].y = vrow[VPAD + n * 16 + m];
      }
#else
      const int sk0 = s0 + t * 4 + 2 * half;
      const float* vr0 = (sk0     == cur) ? vcur : (vbase + (long)sk0 * HDIM);
      const float* vr1 = (sk0 + 1 == cur) ? vcur : (vbase + (long)(sk0 + 1) * HDIM);
#pragma unroll
      for (int n = 0; n < 8; ++n) {
        vb[n].x = vr0[n * 16 + m];
        vb[n].y = vr1[n * 16 + m];
      }
#endif
#pragma unroll
      for (int n = 0; n < 8; ++n)
        ctx[n] = wmma4(pa[t], vb[n], ctx[n]);
    }
  }

  // ---- merge the 8 wave-partials (rescale to global max, sum, normalize)
  if (half == 0) {
#pragma unroll
    for (int i = 0; i < GQ; ++i) {
#pragma unroll
      for (int n = 0; n < 8; ++n) lds_ctx[wave][i][n * 16 + m] = ctx[n][i];
      if (m == 0) { lds_m[wave][i] = mrow[i]; lds_l[wave][i] = lrow[i]; }
    }
  }
  __syncthreads();

  for (int idx = tid; idx < GQ * HDIM; idx += blockDim.x) {
    const int g = idx / HDIM;
    const int d = idx % HDIM;
    float gm = -1e30f;
#pragma unroll
    for (int w = 0; w < 8; ++w) gm = fmaxf(gm, lds_m[w][g]);
    float L = 0.f, V = 0.f;
#pragma unroll
    for (int w = 0; w < 8; ++w) {
      const float e = __expf(lds_m[w][g] - gm);
      L += lds_l[w][g] * e;
      V += lds_ctx[w][g][d] * e;
    }
    ctxout[(long)(b * NHEAD + kv * GQ + g) * HDIM + d] = V / L;
  }
}

// ---------------------------------------------------------------------------
extern "C" void kernel_launch(void* const* d_in, const int* in_sizes, int n_in,
                              void* d_out, int out_size, void* d_ws, size_t ws_size,
                              hipStream_t stream) {
  (void)in_sizes; (void)n_in; (void)out_size; (void)ws_size;
  const float* x       = (const float*)d_in[0];   // (1,1,32,4096)
  const float* wqkv    = (const float*)d_in[1];   // (4096,6144)
  const float* wo      = (const float*)d_in[2];   // (4096,4096)
  const float* rot     = (const float*)d_in[3];   // (128,128)
  const float* cache_k = (const float*)d_in[4];   // (32,8,4096,128)
  const float* cache_v = (const float*)d_in[5];   // (32,8,4096,128)
  const float* mask    = (const float*)d_in[6];   // (1,32,32,4096)
  const int*   curpos  = (const int*)d_in[7];     // scalar

  float* ws   = (float*)d_ws;
  float* xqkv = ws;                                  // 32*6144
  float* qrot = xqkv + (long)BATCH * QKVN;           // 32*32*128
  float* krot = qrot + (long)BATCH * NHEAD * HDIM;   // 32*8*128
  float* ctxb = krot + (long)BATCH * NKVH * HDIM;    // 32*32*128

  // 1) fused QKV projection: xqkv = x @ wqkv   (M=32 -> MT=2, N=6144, K=4096)
  gemm_f32_wmma<2><<<dim3((QKVN / 16) / 4, 1), 128, 0, stream>>>(
      x, 0, wqkv, xqkv, 0, BATCH, QKVN, DMODEL, DMODEL, QKVN, QKVN);

  // 2) rotary on Q: per batch, (32x128) @ rot(128x128)
  gemm_f32_wmma<2><<<dim3((HDIM / 16) / 4, BATCH), 128, 0, stream>>>(
      xqkv, QKVN, rot, qrot, (long)NHEAD * HDIM,
      NHEAD, HDIM, HDIM, HDIM, HDIM, HDIM);

  // 3) rotary on K: per batch, (8x128) @ rot(128x128)  (M padded inside tile)
  gemm_f32_wmma<1><<<dim3((HDIM / 16) / 4, BATCH), 128, 0, stream>>>(
      xqkv + NHEAD * HDIM, QKVN, rot, krot, (long)NKVH * HDIM,
      NKVH, HDIM, HDIM, HDIM, HDIM, HDIM);

  // 4) flash attention over the 4096-token window (streams the 1 GB KV cache)
  attn_flash_wmma<<<dim3(BATCH * NKVH), 256, 0, stream>>>(
      qrot, krot, xqkv, cache_k, cache_v, mask, curpos, ctxb);

  // 5) output projection: out = ctx @ wo  (M=32 -> MT=2, N=4096, K=4096)
  gemm_f32_wmma<2><<<dim3((DMODEL / 16) / 4, 1), 128, 0, stream>>>(
      ctxb, 0, wo, (float*)d_out, 0, BATCH, DMODEL, DMODEL, DMODEL, DMODEL, DMODEL);
}